// ConvOffset2d_31576599560311
// MI455X (gfx1250) — compile-verified
//
#include <hip/hip_runtime.h>
#include <hip/hip_bf16.h>

typedef __attribute__((ext_vector_type(2))) float v2f;
typedef __attribute__((ext_vector_type(8))) float v8f;

#define B_    4
#define CIN_  64
#define H_    128
#define W_    128
#define COUT_ 64
#define KSZ_  3
#define DG_   8
#define CG_   8
#define KK_   9
#define HO_   128
#define WO_   128
#define SLAB_ 72          // CG_*KK_ contiguous contraction indices per deformable group
#define CSTR_ 132         // padded N-stride for cols tile (breaks LDS bank conflicts)

__global__ __launch_bounds__(256)
void dcn_wmma_f32_kernel(const float* __restrict__ xin,
                         const float* __restrict__ offset,
                         const float* __restrict__ weight,
                         const float* __restrict__ bias,
                         float* __restrict__ out)
{
    // LDS: cols[k=0..71][n=0..127] (stride 132) + transposed weight slab wT[k][m]
    __shared__ float cols[SLAB_ * CSTR_];   // 38016 B
    __shared__ float wT[SLAB_ * COUT_];     // 18432 B   (total 56448 B)

    const int y    = blockIdx.x;            // output row
    const int b    = blockIdx.y;            // batch
    const int tid  = threadIdx.x;
    const int lane = tid & 31;
    const int wv   = tid >> 5;              // wave 0..7
    const int mo   = wv & 3;                // M tile (16 rows of COUT)
    const int nb   = (wv >> 2) * 4;         // first of 4 owned N tiles

    const int l15   = lane & 15;
    const int lhalf = lane >> 4;            // 0: K+0,1   1: K+2,3  (32-bit A/B lane split)

    v8f acc[4] = {v8f{}, v8f{}, v8f{}, v8f{}};

    const float gy = (float)y - 1.0f;       // y*STRIDE - PAD

    for (int g = 0; g < DG_; ++g) {
        // ---- stage weight slab, transposed: wT[k*64 + o] = weight[o][g*72 + k] ----
        for (int i = tid; i < COUT_ * SLAB_; i += 256) {
            int o = i / SLAB_;
            int k = i - o * SLAB_;
            wT[k * COUT_ + o] = weight[(size_t)o * (CIN_ * KK_) + g * SLAB_ + k];
        }
        // ---- deformable im2col for this slab: 9 taps x 128 columns ----
        for (int item = tid; item < KK_ * WO_; item += 256) {
            int p  = item >> 7;             // tap 0..8   (WO_==128)
            int xo = item & 127;            // output column
            int kh = p / 3, kw = p - kh * 3;

            const float* offp = offset +
                ((((size_t)b * (2 * DG_ * KK_) + (size_t)(g * KK_ + p) * 2) * HO_ + y) * WO_ + xo);
            float dy = offp[0];
            float dx = offp[(size_t)HO_ * WO_];

            float py = gy + (float)kh + dy;
            float px = (float)xo - 1.0f + (float)kw + dx;
            float fy = floorf(py), fx = floorf(px);
            float ly = py - fy,    lx = px - fx;
            int y0 = (int)fy, x0 = (int)fx;
            int y1 = y0 + 1,  x1 = x0 + 1;

            bool vy0 = (y0 >= 0) && (y0 < H_);
            bool vy1 = (y1 >= 0) && (y1 < H_);
            bool vx0 = (x0 >= 0) && (x0 < W_);
            bool vx1 = (x1 >= 0) && (x1 < W_);

            float w00 = (vy0 && vx0) ? (1.0f - ly) * (1.0f - lx) : 0.0f;
            float w01 = (vy0 && vx1) ? (1.0f - ly) * lx          : 0.0f;
            float w10 = (vy1 && vx0) ? ly * (1.0f - lx)          : 0.0f;
            float w11 = (vy1 && vx1) ? ly * lx                   : 0.0f;

            int cy0 = min(max(y0, 0), H_ - 1);
            int cy1 = min(max(y1, 0), H_ - 1);
            int cx0 = min(max(x0, 0), W_ - 1);
            int cx1 = min(max(x1, 0), W_ - 1);
            int i00 = cy0 * W_ + cx0;
            int i01 = cy0 * W_ + cx1;
            int i10 = cy1 * W_ + cx0;
            int i11 = cy1 * W_ + cx1;

            const float* xb = xin + ((size_t)b * CIN_ + g * CG_) * (H_ * W_);
            #pragma unroll
            for (int c = 0; c < CG_; ++c) {
                const float* xc = xb + (size_t)c * (H_ * W_);
                float v = w00 * xc[i00] + w01 * xc[i01] + w10 * xc[i10] + w11 * xc[i11];
                cols[(c * KK_ + p) * CSTR_ + xo] = v;
            }
        }
        __syncthreads();

        // ---- GEMM slab: 18 k-steps of V_WMMA_F32_16X16X4_F32 ----
        const int am = mo * 16 + l15;       // A-matrix row for this lane
        for (int kk = 0; kk < SLAB_; kk += 4) {
            int ka = kk + lhalf * 2;        // low lanes take K+0,1; high lanes K+2,3
            v2f a;
            a.x = wT[ka * COUT_ + am];
            a.y = wT[(ka + 1) * COUT_ + am];
            #pragma unroll
            for (int j = 0; j < 4; ++j) {
                int n = (nb + j) * 16 + l15;
                v2f bf;
                bf.x = cols[ka * CSTR_ + n];
                bf.y = cols[(ka + 1) * CSTR_ + n];
                acc[j] = __builtin_amdgcn_wmma_f32_16x16x4_f32(
                    /*neg_a=*/false, a, /*neg_b=*/false, bf,
                    /*c_mod=*/(short)0, acc[j],
                    /*reuse_a=*/false, /*reuse_b=*/false);
            }
        }
        __syncthreads();    // cols/wT rewritten next slab
    }

    // ---- epilogue: bias + store; C/D layout: VGPR r -> M=r (+8 for high lanes) ----
    #pragma unroll
    for (int j = 0; j < 4; ++j) {
        int n = (nb + j) * 16 + l15;
        #pragma unroll
        for (int r = 0; r < 8; ++r) {
            int m = mo * 16 + lhalf * 8 + r;
            out[(((size_t)b * COUT_ + m) * HO_ + y) * WO_ + n] = acc[j][r] + bias[m];
        }
    }
}

extern "C" void kernel_launch(void* const* d_in, const int* in_sizes, int n_in,
                              void* d_out, int out_size, void* d_ws, size_t ws_size,
                              hipStream_t stream) {
    const float* x      = (const float*)d_in[0];
    const float* offset = (const float*)d_in[1];
    const float* weight = (const float*)d_in[2];
    const float* bias   = (const float*)d_in[3];
    float* out          = (float*)d_out;

    dim3 grid(HO_, B_);     // one workgroup per (output row, batch)
    dim3 block(256);        // 8 waves
    dcn_wmma_f32_kernel<<<grid, block, 0, stream>>>(x, offset, weight, bias, out);
}